// VGAEModel_68427418960020
// MI455X (gfx1250) — compile-verified
//
#include <hip/hip_runtime.h>
#include <hip/hip_bf16.h>
#include <math.h>

#define N_NODES 100000
#define N_EDGES 1600000
#define DIM_IN  128
#define DIM_HID 64
#define DIM_OUT 32

typedef __attribute__((ext_vector_type(2))) float f32x2;
typedef __attribute__((ext_vector_type(8))) float f32x8;

// ---------------- degree / normalization ----------------

__global__ void k_fill(float* __restrict__ p, float v, int n) {
    int i = blockIdx.x * blockDim.x + threadIdx.x;
    if (i < n) p[i] = v;
}

__global__ void k_count_deg(const int* __restrict__ dst, float* __restrict__ deg, int e) {
    int i = blockIdx.x * blockDim.x + threadIdx.x;
    if (i < e) atomicAdd(&deg[dst[i]], 1.0f);
}

__global__ void k_rsqrt_inplace(float* __restrict__ p, int n) {
    int i = blockIdx.x * blockDim.x + threadIdx.x;
    if (i < n) p[i] = rsqrtf(p[i]);   // deg >= 1 always (self loop)
}

// ---------------- WMMA fp32 GEMM with async LDS weight staging ----------------
// C[M x NC] = A[M x K] @ W[K x NC], all row-major. M%16==0; K, NC compile-time
// (so LDS/global offsets fold into immediate offsets). One wave per 16x16 tile.
// W staged into LDS once per block via GLOBAL_LOAD_ASYNC_TO_LDS_B128 (ASYNCcnt),
// consumed through ds_load for the WMMA B fragments.
// Fragment layouts per CDNA5 ISA 7.12.2:
//   A (16x4, 2 VGPR): lanes 0-15 -> K={k,k+1}, lanes 16-31 -> K={k+2,k+3}, M=lane%16
//   B (4x16, 2 VGPR): VGPR v holds rows v (lanes 0-15) and v+2 (lanes 16-31), N=lane%16
//   C (16x16, 8 VGPR): VGPR v holds M=v (lanes 0-15) and M=v+8 (lanes 16-31), N=lane%16
template <int K, int NC>
__global__ void k_gemm_wmma_f32(const float* __restrict__ A,
                                const float* __restrict__ W,
                                float* __restrict__ C,
                                int M) {
    __shared__ float sW[K * NC];

    // ---- stage W into LDS with async copies (16 B per lane per issue) ----
    {
        const int tid = threadIdx.x;
        constexpr int totalVec4 = (K * NC) >> 2;          // multiple of 256 for our shapes
        const unsigned ldsBase = (unsigned)(uintptr_t)(void*)sW;   // LDS byte offset
#pragma unroll
        for (int i = tid; i < totalVec4; i += 256) {
            const float* g = W + 4 * i;                   // 16B-aligned global source
            unsigned l = ldsBase + 16u * i;               // LDS destination offset
            asm volatile("global_load_async_to_lds_b128 %0, %1, off"
                         :: "v"(l), "v"(g) : "memory");
        }
        asm volatile("s_wait_asynccnt 0x0" ::: "memory");
    }
    __syncthreads();

    const int wavesPerBlock = blockDim.x >> 5;
    const int wid  = threadIdx.x >> 5;
    const int lane = threadIdx.x & 31;
    constexpr int ncTiles = NC >> 4;
    const int gid = blockIdx.x * wavesPerBlock + wid;     // wave-uniform
    const int totalTiles = (M >> 4) * ncTiles;

    if (gid < totalTiles) {                               // whole-wave branch: EXEC all-1s inside
        const int tM = gid / ncTiles;
        const int tN = gid - tM * ncTiles;
        const int half = lane >> 4;                       // 0: lanes 0-15, 1: lanes 16-31
        const int m    = lane & 15;
        const int col  = tN * 16 + m;

        const float* __restrict__ arow = A + (size_t)(tM * 16 + m) * (size_t)K + 2 * half;
        const float* __restrict__ brow = sW + 2 * half * NC + col;

        f32x8 c = {0.f, 0.f, 0.f, 0.f, 0.f, 0.f, 0.f, 0.f};

#pragma unroll
        for (int k = 0; k < K; k += 4) {
            f32x2 a, b;
            a.x = arow[k];                                // global_load, imm offset
            a.y = arow[k + 1];
            b.x = brow[k * NC];                           // ds_load, imm offset
            b.y = brow[k * NC + NC];
            c = __builtin_amdgcn_wmma_f32_16x16x4_f32(
                    /*neg_a=*/false, a, /*neg_b=*/false, b,
                    /*c_mod=*/(short)0, c, /*reuse_a=*/false, /*reuse_b=*/false);
        }

        float* __restrict__ crow = C + (size_t)(tM * 16 + half * 8) * (size_t)NC + col;
#pragma unroll
        for (int v = 0; v < 8; ++v)
            crow[v * NC] = c[v];
    }
}

// ---------------- GCN conv 1 (HID=64) ----------------

// acc[i][k] = b1[k] + h0[i][k] * dis[i]^2   (bias + self-loop term), float4-wide
__global__ void k_init_acc64(const float4* __restrict__ h0, const float* __restrict__ dis,
                             const float4* __restrict__ bias, float4* __restrict__ acc, int n) {
    int i = blockIdx.x * blockDim.x + threadIdx.x;   // over n*16 float4s
    if (i < n * (DIM_HID / 4)) {
        int node = i >> 4;
        int k4   = i & 15;
        float d  = dis[node];
        float d2 = d * d;
        float4 h = h0[i];
        float4 b = bias[k4];
        float4 o;
        o.x = b.x + h.x * d2; o.y = b.y + h.y * d2;
        o.z = b.z + h.z * d2; o.w = b.w + h.w * d2;
        acc[i] = o;
    }
}

// edge scatter: acc[dst][k..k+3] += h0[src][k..k+3] * dis[src]*dis[dst]
__global__ void k_scatter64(const int* __restrict__ src, const int* __restrict__ dst,
                            const float* __restrict__ dis, const float4* __restrict__ h0,
                            float* __restrict__ acc) {
    int t = blockIdx.x * blockDim.x + threadIdx.x;   // over E*16 (fits in int)
    int e = t >> 4;
    int k4 = t & 15;
    if (e < N_EDGES) {
        int s = src[e], d = dst[e];
        float nrm = dis[s] * dis[d];
        float4 h = h0[(size_t)s * (DIM_HID / 4) + k4];
        float* a = acc + (size_t)d * DIM_HID + 4 * k4;
        atomicAdd(a + 0, h.x * nrm);
        atomicAdd(a + 1, h.y * nrm);
        atomicAdd(a + 2, h.z * nrm);
        atomicAdd(a + 3, h.w * nrm);
    }
}

// relu + row-wise L2 normalize, one wave (32 lanes) per 64-wide row
__global__ void k_relu_l2norm64(float* __restrict__ h, int n) {
    int wid  = threadIdx.x >> 5;
    int lane = threadIdx.x & 31;
    int row  = blockIdx.x * (blockDim.x >> 5) + wid;
    if (row >= n) return;
    float* r = h + (size_t)row * DIM_HID;
    float v0 = fmaxf(r[lane], 0.0f);
    float v1 = fmaxf(r[lane + 32], 0.0f);
    float s = v0 * v0 + v1 * v1;
#pragma unroll
    for (int off = 16; off > 0; off >>= 1)
        s += __shfl_xor(s, off, 32);
    float inv = 1.0f / fmaxf(sqrtf(s), 1e-12f);
    r[lane]      = v0 * inv;
    r[lane + 32] = v1 * inv;
}

// ---------------- GCN conv 2/3 (OUT=32), fused mu + logstd ----------------

__global__ void k_init_acc32x2(const float4* __restrict__ mu0, const float4* __restrict__ ls0,
                               const float* __restrict__ dis,
                               const float4* __restrict__ bmu, const float4* __restrict__ bls,
                               float4* __restrict__ mu_acc, float4* __restrict__ ls_acc, int n) {
    int i = blockIdx.x * blockDim.x + threadIdx.x;   // over n*8 float4s
    if (i < n * (DIM_OUT / 4)) {
        int node = i >> 3;
        int k4   = i & 7;
        float d2 = dis[node];
        d2 *= d2;
        float4 m = mu0[i], l = ls0[i];
        float4 bm = bmu[k4], bl = bls[k4];
        float4 om, ol;
        om.x = bm.x + m.x * d2; om.y = bm.y + m.y * d2;
        om.z = bm.z + m.z * d2; om.w = bm.w + m.w * d2;
        ol.x = bl.x + l.x * d2; ol.y = bl.y + l.y * d2;
        ol.z = bl.z + l.z * d2; ol.w = bl.w + l.w * d2;
        mu_acc[i] = om;
        ls_acc[i] = ol;
    }
}

__global__ void k_scatter32x2(const int* __restrict__ src, const int* __restrict__ dst,
                              const float* __restrict__ dis,
                              const float4* __restrict__ mu0, const float4* __restrict__ ls0,
                              float* __restrict__ mu_acc, float* __restrict__ ls_acc) {
    int t = blockIdx.x * blockDim.x + threadIdx.x;   // over E*8
    int e = t >> 3;
    int k4 = t & 7;
    if (e < N_EDGES) {
        int s = src[e], d = dst[e];
        float nrm = dis[s] * dis[d];
        size_t si = (size_t)s * (DIM_OUT / 4) + k4;
        float4 m = mu0[si], l = ls0[si];
        float* am = mu_acc + (size_t)d * DIM_OUT + 4 * k4;
        float* al = ls_acc + (size_t)d * DIM_OUT + 4 * k4;
        atomicAdd(am + 0, m.x * nrm);
        atomicAdd(am + 1, m.y * nrm);
        atomicAdd(am + 2, m.z * nrm);
        atomicAdd(am + 3, m.w * nrm);
        atomicAdd(al + 0, l.x * nrm);
        atomicAdd(al + 1, l.y * nrm);
        atomicAdd(al + 2, l.z * nrm);
        atomicAdd(al + 3, l.w * nrm);
    }
}

// ---------------- reparametrize + decode ----------------

__global__ void k_reparam(const float* __restrict__ mu, const float* __restrict__ ls,
                          const float* __restrict__ eps, float* __restrict__ z, int n) {
    int i = blockIdx.x * blockDim.x + threadIdx.x;   // over n*32
    if (i < n * DIM_OUT) {
        float l = fminf(fmaxf(ls[i], -10.0f), 10.0f);
        z[i] = mu[i] + eps[i] * __expf(l);
    }
}

// one wave per edge: adj[e] = sigmoid(<z[src], z[dst]>)
__global__ void k_decode(const int* __restrict__ src, const int* __restrict__ dst,
                         const float* __restrict__ z, float* __restrict__ adj) {
    int wid  = threadIdx.x >> 5;
    int lane = threadIdx.x & 31;
    int e = blockIdx.x * (blockDim.x >> 5) + wid;
    if (e >= N_EDGES) return;
    int s = src[e], d = dst[e];
    float p = z[(size_t)s * DIM_OUT + lane] * z[(size_t)d * DIM_OUT + lane];
#pragma unroll
    for (int off = 16; off > 0; off >>= 1)
        p += __shfl_xor(p, off, 32);
    if (lane == 0)
        adj[e] = 1.0f / (1.0f + __expf(-p));
}

// ---------------- host launcher ----------------

extern "C" void kernel_launch(void* const* d_in, const int* in_sizes, int n_in,
                              void* d_out, int out_size, void* d_ws, size_t ws_size,
                              hipStream_t stream) {
    (void)in_sizes; (void)n_in; (void)out_size; (void)ws_size;

    const float* x    = (const float*)d_in[0];
    const int*   ei   = (const int*)d_in[1];
    const float* eps  = (const float*)d_in[2];
    const float* W1   = (const float*)d_in[3];
    const float* b1   = (const float*)d_in[4];
    const float* Wmu  = (const float*)d_in[5];
    const float* bmu  = (const float*)d_in[6];
    const float* Wls  = (const float*)d_in[7];
    const float* bls  = (const float*)d_in[8];

    const int* src = ei;
    const int* dst = ei + N_EDGES;

    float* adj    = (float*)d_out;                          // [E]
    float* mu_out = adj + N_EDGES;                          // [N,32]
    float* ls_out = mu_out + (size_t)N_NODES * DIM_OUT;     // [N,32]

    float* ws   = (float*)d_ws;
    float* dis  = ws;                                       // [N]
    float* bufA = ws + N_NODES;                             // [N*64]  h0, then mu0/ls0
    float* bufB = bufA + (size_t)N_NODES * DIM_HID;         // [N*64]  h,  then z
    float* h0  = bufA;
    float* h   = bufB;
    float* mu0 = bufA;                                      // reuse of h0 region
    float* ls0 = bufA + (size_t)N_NODES * DIM_OUT;
    float* z   = bufB;                                      // reuse of h region

    const int B = 256;

    // 1) degrees: dis = 1 (self loop) + count(dst) ; then dis = rsqrt(deg)
    k_fill<<<(N_NODES + B - 1) / B, B, 0, stream>>>(dis, 1.0f, N_NODES);
    k_count_deg<<<(N_EDGES + B - 1) / B, B, 0, stream>>>(dst, dis, N_EDGES);
    k_rsqrt_inplace<<<(N_NODES + B - 1) / B, B, 0, stream>>>(dis, N_NODES);

    // 2) h0 = x @ W1   (100000x128 @ 128x64) via WMMA f32 + async LDS weights
    {
        int tiles = (N_NODES / 16) * (DIM_HID / 16);
        k_gemm_wmma_f32<DIM_IN, DIM_HID><<<(tiles + 7) / 8, B, 0, stream>>>(x, W1, h0, N_NODES);
    }

    // 3) conv1 aggregate: init (bias + self-loop) then edge scatter; relu + L2 norm
    k_init_acc64<<<((N_NODES * (DIM_HID / 4)) + B - 1) / B, B, 0, stream>>>(
        (const float4*)h0, dis, (const float4*)b1, (float4*)h, N_NODES);
    k_scatter64<<<((N_EDGES * 16) + B - 1) / B, B, 0, stream>>>(
        src, dst, dis, (const float4*)h0, h);
    k_relu_l2norm64<<<(N_NODES + 7) / 8, B, 0, stream>>>(h, N_NODES);

    // 4) mu0 = h @ Wmu, ls0 = h @ Wls  (100000x64 @ 64x32) via WMMA f32
    {
        int tiles = (N_NODES / 16) * (DIM_OUT / 16);
        k_gemm_wmma_f32<DIM_HID, DIM_OUT><<<(tiles + 7) / 8, B, 0, stream>>>(h, Wmu, mu0, N_NODES);
        k_gemm_wmma_f32<DIM_HID, DIM_OUT><<<(tiles + 7) / 8, B, 0, stream>>>(h, Wls, ls0, N_NODES);
    }

    // 5) conv2/3 aggregate directly into d_out mu/logstd regions
    k_init_acc32x2<<<((N_NODES * (DIM_OUT / 4)) + B - 1) / B, B, 0, stream>>>(
        (const float4*)mu0, (const float4*)ls0, dis,
        (const float4*)bmu, (const float4*)bls,
        (float4*)mu_out, (float4*)ls_out, N_NODES);
    k_scatter32x2<<<((N_EDGES * 8) + B - 1) / B, B, 0, stream>>>(
        src, dst, dis, (const float4*)mu0, (const float4*)ls0, mu_out, ls_out);

    // 6) z = mu + eps * exp(clamp(logstd, -10, 10))
    k_reparam<<<((N_NODES * DIM_OUT) + B - 1) / B, B, 0, stream>>>(mu_out, ls_out, eps, z, N_NODES);

    // 7) adj_pred = sigmoid(<z[src], z[dst]>), one wave per edge
    k_decode<<<(N_EDGES + 7) / 8, B, 0, stream>>>(src, dst, z, adj);
}